// HMPGCNConv_11278584119445
// MI455X (gfx1250) — compile-verified
//
#include <hip/hip_runtime.h>
#include <hip/hip_bf16.h>

namespace {
constexpr int   kN       = 100000;
constexpr int   kE       = 1000000;
constexpr int   kD       = 128;
constexpr float kMinNorm = 1e-15f;
constexpr float kMaxNorm = 0.996f;   // (1 - EPS)/sqrt(c), EPS=4e-3, c=1
constexpr float kSlope   = 0.01f;
}

typedef float v2f __attribute__((ext_vector_type(2)));
typedef float v8f __attribute__((ext_vector_type(8)));

__device__ __forceinline__ float wave_sum(float v) {
  v += __shfl_xor(v, 1, 32);
  v += __shfl_xor(v, 2, 32);
  v += __shfl_xor(v, 4, 32);
  v += __shfl_xor(v, 8, 32);
  v += __shfl_xor(v, 16, 32);
  return v;
}

__device__ __forceinline__ float dot4(const float4 a, const float4 b) {
  return a.x * b.x + a.y * b.y + a.z * b.z + a.w * b.w;
}
__device__ __forceinline__ float4 scale4(const float4 a, float s) {
  float4 r; r.x = a.x * s; r.y = a.y * s; r.z = a.z * s; r.w = a.w * s; return r;
}
__device__ __forceinline__ float4 axpby4(float a, const float4 x, float b, const float4 y) {
  float4 r;
  r.x = a * x.x + b * y.x; r.y = a * x.y + b * y.y;
  r.z = a * x.z + b * y.z; r.w = a * x.w + b * y.w;
  return r;
}

// ---------------- hyp_bias = expmap0(bias), one wave ----------------
__global__ void bias_expmap0(const float* __restrict__ bias, float* __restrict__ hb) {
  const int lane = threadIdx.x & 31;
  float4 u = *(const float4*)(bias + lane * 4);
  float n2 = wave_sum(dot4(u, u));
  float un = fmaxf(sqrtf(n2), kMinNorm);
  float sc = tanhf(un) / un;
  *(float4*)(hb + lane * 4) = scale4(u, sc);
}

// ---------------- degree / norm ----------------
__global__ void fill_deg(float* __restrict__ deg) {
  int i = blockIdx.x * blockDim.x + threadIdx.x;
  if (i < kN) deg[i] = 1.0f;  // self loop
}

__global__ void deg_accum(const int* __restrict__ ei, float* __restrict__ deg) {
  int e = blockIdx.x * blockDim.x + threadIdx.x;
  if (e < kE) unsafeAtomicAdd(&deg[ei[e]], 1.0f);
}

__global__ void deg_to_dinv(float* __restrict__ deg) {
  int i = blockIdx.x * blockDim.x + threadIdx.x;
  if (i < kN) {
    float d = deg[i];
    deg[i] = (d > 0.0f) ? rsqrtf(d) : 0.0f;
  }
}

// ---------------- h = x @ W.T via V_WMMA_F32_16X16X4_F32 ----------------
// grid (kN/16, 2); block = 128 (4 waves). Tile: 16 nodes x 64 outputs.
#define LDS_STRIDE 132  // 132 % 64 == 4 -> conflict-free fragment reads

__global__ __launch_bounds__(128) void gemm_wmma(const float* __restrict__ x,
                                                 const float* __restrict__ W,
                                                 float* __restrict__ h) {
  __shared__ float xs[16 * LDS_STRIDE];
  __shared__ float wsh[64 * LDS_STRIDE];

  const int tid      = threadIdx.x;
  const int nodeBase = blockIdx.x * 16;
  const int outHalf  = blockIdx.y;   // 0 or 1 -> outputs [0,64) or [64,128)

  // x tile: 16x128 f32 = 512 float4, 4 per thread, coalesced
  for (int i = tid; i < 512; i += 128) {
    int m = i >> 5, k4 = (i & 31) << 2;
    float4 v = *(const float4*)(x + (size_t)(nodeBase + m) * kD + k4);
    *(float4*)&xs[m * LDS_STRIDE + k4] = v;
  }
  // W half: 64x128 f32 = 2048 float4, 16 per thread (W is [DOUT][DIN] = o-major)
  for (int i = tid; i < 2048; i += 128) {
    int o = i >> 5, k4 = (i & 31) << 2;
    float4 v = *(const float4*)(W + (size_t)(outHalf * 64 + o) * kD + k4);
    *(float4*)&wsh[o * LDS_STRIDE + k4] = v;
  }
  __syncthreads();

  const int lane = tid & 31;
  const int wave = tid >> 5;            // 0..3
  const int m    = lane & 15;           // A row / B column within tile
  const int koff = (lane >> 4) << 1;    // lanes 16-31 hold K+2, K+3

  const float* ax = &xs[m * LDS_STRIDE + koff];
  const float* bx = &wsh[(wave * 16 + m) * LDS_STRIDE + koff];

  v8f acc = {0.f, 0.f, 0.f, 0.f, 0.f, 0.f, 0.f, 0.f};
#pragma unroll
  for (int kk = 0; kk < kD; kk += 4) {
    v2f a = *(const v2f*)(ax + kk);   // A[m][kk+koff .. +1]
    v2f b = *(const v2f*)(bx + kk);   // B[kk+koff..][n] = W[o][kk+koff..]
    acc = __builtin_amdgcn_wmma_f32_16x16x4_f32(false, a, false, b,
                                                (short)0, acc, false, false);
  }

  // D layout: VGPR v -> M = v + (lane>=16 ? 8 : 0), N = lane & 15
  const int nodeOff = (lane >> 4) << 3;
  const int outc    = outHalf * 64 + wave * 16 + (lane & 15);
#pragma unroll
  for (int v = 0; v < 8; ++v)
    h[(size_t)(nodeBase + v + nodeOff) * kD + outc] = acc[v];
}

// ---------------- per-node hyperbolic chain: h -> s, lamb ----------------
// one wave per node, 4 floats/lane
__global__ __launch_bounds__(256) void hyper_pointwise(const float* __restrict__ h,
                                                       const float* __restrict__ hb,
                                                       float* __restrict__ s,
                                                       float* __restrict__ lamb) {
  const int lane = threadIdx.x & 31;
  const int node = blockIdx.x * 8 + (threadIdx.x >> 5);
  if (node >= kN) return;

  float4 hv = *(const float4*)(h + (size_t)node * kD + lane * 4);

  // proj
  float n2  = wave_sum(dot4(hv, hv));
  float nrm = fmaxf(sqrtf(n2), kMinNorm);
  if (nrm > kMaxNorm) hv = scale4(hv, kMaxNorm / nrm);

  // mobius_add(h, hyp_bias)
  float4 bv = *(const float4*)(hb + lane * 4);
  float x2 = wave_sum(dot4(hv, hv));
  float y2 = wave_sum(dot4(bv, bv));
  float xy = wave_sum(dot4(hv, bv));
  float ca  = 1.0f + 2.0f * xy + y2;
  float cb  = 1.0f - x2;
  float den = fmaxf(1.0f + 2.0f * xy + x2 * y2, kMinNorm);
  hv = scale4(axpby4(ca, hv, cb, bv), 1.0f / den);

  // proj
  n2  = wave_sum(dot4(hv, hv));
  nrm = fmaxf(sqrtf(n2), kMinNorm);
  if (nrm > kMaxNorm) hv = scale4(hv, kMaxNorm / nrm);

  // p2k
  n2 = wave_sum(dot4(hv, hv));
  float4 sv = scale4(hv, 2.0f / (1.0f + n2));

  // lorenz factor
  float sn2 = wave_sum(dot4(sv, sv));
  float lam = rsqrtf(fmaxf(1.0f - sn2, kMinNorm));

  *(float4*)(s + (size_t)node * kD + lane * 4) = sv;
  if (lane == 0) lamb[node] = lam;
}

// ---------------- self-loop contribution initializes s_out / tmp ----------------
__global__ __launch_bounds__(256) void agg_init(const float* __restrict__ s,
                                                const float* __restrict__ lamb,
                                                const float* __restrict__ dinv,
                                                float* __restrict__ sout,
                                                float* __restrict__ tmp) {
  const int lane = threadIdx.x & 31;
  const int node = blockIdx.x * 8 + (threadIdx.x >> 5);
  if (node >= kN) return;
  float di = dinv[node];
  float w  = di * di * lamb[node];
  float4 sv = *(const float4*)(s + (size_t)node * kD + lane * 4);
  *(float4*)(sout + (size_t)node * kD + lane * 4) = scale4(sv, w);
  if (lane == 0) tmp[node] = w;
}

// ---------------- edge scatter: one wave per edge ----------------
__global__ __launch_bounds__(256) void agg_edges(const int* __restrict__ ei,
                                                 const float* __restrict__ s,
                                                 const float* __restrict__ lamb,
                                                 const float* __restrict__ dinv,
                                                 float* __restrict__ sout,
                                                 float* __restrict__ tmp) {
  const int lane = threadIdx.x & 31;
  const long e = (long)blockIdx.x * 8 + (threadIdx.x >> 5);
  if (e >= kE) return;
  const int r = ei[e];
  const int c = ei[kE + e];
  const float w = dinv[r] * dinv[c] * lamb[c];
  float4 sv = *(const float4*)(s + (size_t)c * kD + lane * 4);
  float* dst = sout + (size_t)r * kD + lane * 4;
  unsafeAtomicAdd(dst + 0, w * sv.x);
  unsafeAtomicAdd(dst + 1, w * sv.y);
  unsafeAtomicAdd(dst + 2, w * sv.z);
  unsafeAtomicAdd(dst + 3, w * sv.w);
  if (lane == 0) unsafeAtomicAdd(tmp + r, w);
}

// ---------------- finalize: divide, k2p, leaky_relu ----------------
__global__ __launch_bounds__(256) void finalize(const float* __restrict__ sout,
                                                const float* __restrict__ tmp,
                                                float* __restrict__ out) {
  const int lane = threadIdx.x & 31;
  const int node = blockIdx.x * 8 + (threadIdx.x >> 5);
  if (node >= kN) return;
  float4 v = *(const float4*)(sout + (size_t)node * kD + lane * 4);
  v = scale4(v, 1.0f / tmp[node]);
  float n2 = wave_sum(dot4(v, v));
  float inv = 1.0f / (1.0f + sqrtf(fmaxf(1.0f - n2, kMinNorm)));
  float4 o = scale4(v, inv);
  o.x = (o.x >= 0.0f) ? o.x : kSlope * o.x;
  o.y = (o.y >= 0.0f) ? o.y : kSlope * o.y;
  o.z = (o.z >= 0.0f) ? o.z : kSlope * o.z;
  o.w = (o.w >= 0.0f) ? o.w : kSlope * o.w;
  *(float4*)(out + (size_t)node * kD + lane * 4) = o;
}

extern "C" void kernel_launch(void* const* d_in, const int* in_sizes, int n_in,
                              void* d_out, int out_size, void* d_ws, size_t ws_size,
                              hipStream_t stream) {
  const float* x    = (const float*)d_in[0];   // [N,128]
  const float* W    = (const float*)d_in[1];   // [128,128]
  const float* bias = (const float*)d_in[2];   // [1,128]
  const int*   ei   = (const int*)d_in[3];     // [2,E]

  const size_t ND = (size_t)kN * kD;
  float* wsf  = (float*)d_ws;
  float* h    = wsf;             // [N,128]  (aliased as s_out after hyper_pointwise)
  float* s    = wsf + ND;        // [N,128]
  float* lamb = wsf + 2 * ND;    // [N]
  float* dinv = lamb + kN;       // [N]      (deg during accumulation)
  float* tmp  = dinv + kN;       // [N]
  float* hb   = tmp + kN;        // [128]
  float* sout = h;               // reuse: h is dead after hyper_pointwise

  bias_expmap0<<<1, 32, 0, stream>>>(bias, hb);

  fill_deg<<<(kN + 255) / 256, 256, 0, stream>>>(dinv);
  deg_accum<<<(kE + 255) / 256, 256, 0, stream>>>(ei, dinv);
  deg_to_dinv<<<(kN + 255) / 256, 256, 0, stream>>>(dinv);

  gemm_wmma<<<dim3(kN / 16, 2), 128, 0, stream>>>(x, W, h);

  hyper_pointwise<<<(kN + 7) / 8, 256, 0, stream>>>(h, hb, s, lamb);

  agg_init<<<(kN + 7) / 8, 256, 0, stream>>>(s, lamb, dinv, sout, tmp);
  agg_edges<<<(kE + 7) / 8, 256, 0, stream>>>(ei, s, lamb, dinv, sout, tmp);

  finalize<<<(kN + 7) / 8, 256, 0, stream>>>(sout, tmp, (float*)d_out);
}